// ContrastiveLoss_37237366456708
// MI455X (gfx1250) — compile-verified
//
#include <hip/hip_runtime.h>
#include <math.h>

// ---------------------------------------------------------------------------
// XBM contrastive loss on MI455X (gfx1250, wave32).
// sim = inputs_col(256x512) * inputs_row(65536x512)^T via V_WMMA_F32_16X16X4_F32
// Phase1: tiled f32 WMMA GEMM + per-(row,chunk) pos-sum / top-10 partials
// Phase2: per-row merge (deterministic tree reductions, no float atomics)
// Phase3: mean over rows -> scalar
// ---------------------------------------------------------------------------

typedef __attribute__((ext_vector_type(2))) float v2f;
typedef __attribute__((ext_vector_type(8))) float v8f;

#define B_ROWS     256
#define M_COLS     65536
#define D_DIM      512
#define KC         64          // K-chunk staged in LDS
#define NC         256         // columns handled per workgroup
#define AB_STRIDE  68          // padded LDS row stride (floats), 16B aligned, %64 != 0
#define SIM_STRIDE 257         // padded sim row stride (floats)
#define NEG_TOPK   10
#define N_CHUNKS   (M_COLS / NC)   // 256
#define EPS_F      1e-5f

// ---------------------------------------------------------------------------
__global__ __launch_bounds__(256)
void xbm_phase1(const float* __restrict__ inputs_col,
                const int*   __restrict__ targets_col,
                const float* __restrict__ inputs_row,
                const int*   __restrict__ target_row,
                float*       __restrict__ pos_ws,    // [256][N_CHUNKS]
                float*       __restrict__ topk_ws)   // [256][N_CHUNKS][NEG_TOPK]
{
    __shared__ float lds_a[16 * AB_STRIDE];     //  4.3 KB  A tile [16][KC]
    __shared__ float lds_b[NC * AB_STRIDE];     // 69.6 KB  B tile [NC][KC]
    __shared__ float lds_sim[16 * SIM_STRIDE];  // 16.4 KB  sim tile [16][NC]
    __shared__ int   lds_trow[NC];
    __shared__ int   lds_tcol[16];

    const int chunk  = blockIdx.x;        // 0..255  (M chunk)
    const int rowblk = blockIdx.y;        // 0..15   (B row block)
    const int tid    = threadIdx.x;       // 0..255  (8 waves)
    const int lane   = tid & 31;
    const int wave   = tid >> 5;          // 0..7
    const int half   = lane >> 4;         // K-half select for A/B fragments
    const int l16    = lane & 15;

    const int rowbase = rowblk * 16;
    const int colbase = chunk * NC;

    v8f acc0 = {};   // tile cols [wave*32 +  0 .. +15]
    v8f acc1 = {};   // tile cols [wave*32 + 16 .. +31]

    for (int kc = 0; kc < D_DIM; kc += KC) {
        __syncthreads();
        // -------- stage A: 16 rows x KC floats, one float4 per thread --------
        {
            const int r = tid >> 4;       // 0..15
            const int v = tid & 15;       // float4 index within row
            const float4 a4 = *(const float4*)(inputs_col
                                + (size_t)(rowbase + r) * D_DIM + kc + v * 4);
            float* dst = lds_a + r * AB_STRIDE + v * 4;
            dst[0] = a4.x; dst[1] = a4.y; dst[2] = a4.z; dst[3] = a4.w;
        }
        // -------- stage B: NC rows x KC floats, 16 float4 per thread ---------
        #pragma unroll 4
        for (int i = 0; i < 16; ++i) {
            const int idx = tid + i * 256;   // 0..4095
            const int c   = idx >> 4;        // column 0..255
            const int v   = idx & 15;
            const float4 b4 = *(const float4*)(inputs_row
                                + (size_t)(colbase + c) * D_DIM + kc + v * 4);
            float* dst = lds_b + c * AB_STRIDE + v * 4;
            dst[0] = b4.x; dst[1] = b4.y; dst[2] = b4.z; dst[3] = b4.w;
        }
        __syncthreads();

        // -------- WMMA over staged K-chunk: 16 k-steps x 2 tiles -------------
        const int n0 = wave * 32;
        #pragma unroll 4
        for (int kk = 0; kk < KC; kk += 4) {
            const int k = kk + 2 * half;  // this lane's K pair (even)
            v2f a, b0, b1;
            { const float* p = lds_a + l16 * AB_STRIDE + k;
              a.x  = p[0]; a.y  = p[1]; }
            { const float* p = lds_b + (n0      + l16) * AB_STRIDE + k;
              b0.x = p[0]; b0.y = p[1]; }
            { const float* p = lds_b + (n0 + 16 + l16) * AB_STRIDE + k;
              b1.x = p[0]; b1.y = p[1]; }
            acc0 = __builtin_amdgcn_wmma_f32_16x16x4_f32(
                       false, a, false, b0, (short)0, acc0, false, false);
            acc1 = __builtin_amdgcn_wmma_f32_16x16x4_f32(
                       false, a, false, b1, (short)0, acc1, false, false);
        }
    }

    __syncthreads();
    // -------- scatter sim fragments to LDS: (vgpr r, lane) -> (m, n) ---------
    {
        const int n0 = wave * 32;
        #pragma unroll
        for (int r = 0; r < 8; ++r) {
            const int m = r + 8 * half;
            lds_sim[m * SIM_STRIDE + n0 + l16]      = acc0[r];
            lds_sim[m * SIM_STRIDE + n0 + 16 + l16] = acc1[r];
        }
    }
    if (tid < NC) lds_trow[tid] = target_row[colbase + tid];
    if (tid < 16) lds_tcol[tid] = targets_col[rowbase + tid];
    __syncthreads();

    // -------- per-row epilogue: deterministic serial scan (threads 0..15) ----
    if (tid < 16) {
        const int m  = tid;
        const int tc = lds_tcol[m];
        float pos = 0.0f;
        float top[NEG_TOPK];
        #pragma unroll
        for (int j = 0; j < NEG_TOPK; ++j) top[j] = -INFINITY;

        for (int n = 0; n < NC; ++n) {
            const float s = lds_sim[m * SIM_STRIDE + n];
            if (lds_trow[n] == tc) {
                if (s < 1.0f - EPS_F) pos += 1.0f - s;
            } else if (s > top[NEG_TOPK - 1]) {
                int j = NEG_TOPK - 1;
                while (j > 0 && top[j - 1] < s) { top[j] = top[j - 1]; --j; }
                top[j] = s;
            }
        }
        const int row = rowbase + m;
        pos_ws[row * N_CHUNKS + chunk] = pos;
        float* dst = topk_ws + ((size_t)row * N_CHUNKS + chunk) * NEG_TOPK;
        #pragma unroll
        for (int j = 0; j < NEG_TOPK; ++j) dst[j] = top[j];
    }
}

// ---------------------------------------------------------------------------
__global__ __launch_bounds__(256)
void xbm_phase2(const float* __restrict__ pos_ws,
                const float* __restrict__ topk_ws,
                float*       __restrict__ row_loss)
{
    __shared__ float cand[N_CHUNKS * NEG_TOPK];   // 2560 candidates
    __shared__ float red_v[256];
    __shared__ int   red_i[256];
    __shared__ float s_pos, s_neg;

    const int row = blockIdx.x;
    const int tid = threadIdx.x;

    // Load this row's candidate lists (thread t owns chunk t's 10 values).
    {
        const float* src = topk_ws + ((size_t)row * N_CHUNKS + tid) * NEG_TOPK;
        #pragma unroll
        for (int j = 0; j < NEG_TOPK; ++j) cand[tid * NEG_TOPK + j] = src[j];
    }
    // Deterministic tree sum of pos partials.
    red_v[tid] = pos_ws[row * N_CHUNKS + tid];
    __syncthreads();
    for (int s = 128; s > 0; s >>= 1) {
        if (tid < s) red_v[tid] += red_v[tid + s];
        __syncthreads();
    }
    if (tid == 0) { s_pos = red_v[0]; s_neg = 0.0f; }
    __syncthreads();

    // 10 iterations of parallel argmax (value, then lowest index: deterministic).
    for (int it = 0; it < NEG_TOPK; ++it) {
        float bv = -INFINITY; int bi = -1;
        #pragma unroll
        for (int j = 0; j < NEG_TOPK; ++j) {
            const float v = cand[tid * NEG_TOPK + j];
            if (v > bv) { bv = v; bi = tid * NEG_TOPK + j; }
        }
        red_v[tid] = bv; red_i[tid] = bi;
        __syncthreads();
        for (int s = 128; s > 0; s >>= 1) {
            if (tid < s) {
                const float ov = red_v[tid + s];
                const int   oi = red_i[tid + s];
                if (ov > red_v[tid] || (ov == red_v[tid] && oi >= 0 && oi < red_i[tid])) {
                    red_v[tid] = ov; red_i[tid] = oi;
                }
            }
            __syncthreads();
        }
        if (tid == 0) {
            if (red_i[0] >= 0 && red_v[0] > -INFINITY) {   // finite check
                s_neg += red_v[0];
                cand[red_i[0]] = -INFINITY;
            }
        }
        __syncthreads();
    }
    if (tid == 0) row_loss[row] = s_pos + s_neg;
}

// ---------------------------------------------------------------------------
__global__ __launch_bounds__(256)
void xbm_phase3(const float* __restrict__ row_loss, float* __restrict__ out)
{
    __shared__ float red[256];
    const int tid = threadIdx.x;
    red[tid] = row_loss[tid];
    __syncthreads();
    for (int s = 128; s > 0; s >>= 1) {
        if (tid < s) red[tid] += red[tid + s];
        __syncthreads();
    }
    if (tid == 0) out[0] = red[0] * (1.0f / (float)B_ROWS);
}

// ---------------------------------------------------------------------------
extern "C" void kernel_launch(void* const* d_in, const int* in_sizes, int n_in,
                              void* d_out, int out_size, void* d_ws, size_t ws_size,
                              hipStream_t stream)
{
    const float* inputs_col  = (const float*)d_in[0];   // [256,512]
    const int*   targets_col = (const int*)  d_in[1];   // [256]
    const float* inputs_row  = (const float*)d_in[2];   // [65536,512]
    const int*   target_row  = (const int*)  d_in[3];   // [65536]

    float* ws      = (float*)d_ws;
    float* topk_ws = ws;                                        // 256*256*10 f
    float* pos_ws  = topk_ws + (size_t)B_ROWS * N_CHUNKS * NEG_TOPK; // 256*256 f
    float* rloss   = pos_ws  + (size_t)B_ROWS * N_CHUNKS;           // 256 f

    dim3 g1(N_CHUNKS, B_ROWS / 16);   // 256 x 16 workgroups
    xbm_phase1<<<g1, 256, 0, stream>>>(inputs_col, targets_col,
                                       inputs_row, target_row,
                                       pos_ws, topk_ws);
    xbm_phase2<<<B_ROWS, 256, 0, stream>>>(pos_ws, topk_ws, rloss);
    xbm_phase3<<<1, 256, 0, stream>>>(rloss, (float*)d_out);
}